// Model_28724741276025
// MI455X (gfx1250) — compile-verified
//
#include <hip/hip_runtime.h>

typedef float v2f __attribute__((ext_vector_type(2)));
typedef float v4f __attribute__((ext_vector_type(4)));
typedef float v8f __attribute__((ext_vector_type(8)));

// ---------------------------------------------------------------------------
// ws layout (floats): [0..127] column-sum of mu_N
//                     [128..135] {h+, h-, hc+, hc-, s+, s-, sc+, sc-}
// ---------------------------------------------------------------------------

__global__ void zero_ws_kernel(float* __restrict__ ws) {
    int t = threadIdx.x;
    if (t < 136) ws[t] = 0.0f;
}

// Streaming column sum of mu_N [rows, 128].  One wave reads one full 512B row
// per iteration as 32 x float4 (global_load_b128, non-temporal: 205MB is read
// exactly once and exceeds the 192MB L2), 8 waves per block, grid-stride over
// row groups.  ~205 MB read => ~8.8us at 23.3 TB/s.
__global__ __launch_bounds__(256) void colsum_kernel(const float* __restrict__ mu,
                                                     float* __restrict__ acc,
                                                     int rows) {
    __shared__ v4f smem[256];
    const int tid  = threadIdx.x;
    const int colg = tid & 31;   // which float4 of the row (0..31)
    const int rsub = tid >> 5;   // wave index in block (0..7)

    v4f s = {0.f, 0.f, 0.f, 0.f};
    const int rstride = (int)gridDim.x * 8;
    for (int r = (int)blockIdx.x * 8 + rsub; r < rows; r += rstride) {
        const v4f* p = reinterpret_cast<const v4f*>(mu + (size_t)r * 128 + (size_t)colg * 4);
        v4f v = __builtin_nontemporal_load(p);
        s += v;
    }
    smem[tid] = s;
    __syncthreads();
    if (tid < 32) {
        v4f t = smem[tid];
        #pragma unroll
        for (int g = 1; g < 8; ++g) t += smem[g * 32 + tid];
        atomicAdd(&acc[tid * 4 + 0], t.x);
        atomicAdd(&acc[tid * 4 + 1], t.y);
        atomicAdd(&acc[tid * 4 + 2], t.z);
        atomicAdd(&acc[tid * 4 + 3], t.w);
    }
}

// Sum of relu(x) and relu(-x) for the four edge-weight vectors.
// blockIdx.y selects the array.  relu positive-homogeneity makes the
// 100000x128 outer-product terms collapse to these two scalars each:
//   sum_i relu(h_i * w) = (sum relu(h_i)) * relu(w) + (sum relu(-h_i)) * relu(-w)
__global__ __launch_bounds__(256) void edge_sums_kernel(const float* __restrict__ h,
                                                        const float* __restrict__ hc,
                                                        const float* __restrict__ s,
                                                        const float* __restrict__ sc,
                                                        int n, float* __restrict__ acc) {
    const float* p = (blockIdx.y == 0) ? h : (blockIdx.y == 1) ? hc
                   : (blockIdx.y == 2) ? s : sc;
    __shared__ float redp[256];
    __shared__ float redn[256];
    const int tid = threadIdx.x;
    float pos = 0.f, neg = 0.f;
    for (int i = (int)blockIdx.x * 256 + tid; i < n; i += (int)gridDim.x * 256) {
        float x = p[i];
        pos += fmaxf(x, 0.f);
        neg += fmaxf(-x, 0.f);
    }
    redp[tid] = pos; redn[tid] = neg;
    __syncthreads();
    for (int off = 128; off > 0; off >>= 1) {
        if (tid < off) { redp[tid] += redp[tid + off]; redn[tid] += redn[tid + off]; }
        __syncthreads();
    }
    if (tid == 0) {
        atomicAdd(&acc[128 + 2 * blockIdx.y + 0], redp[0]);
        atomicAdd(&acc[128 + 2 * blockIdx.y + 1], redn[0]);
    }
}

// Fused 5x [1,128]@[128,128] matvec + xi@W10 + relu via V_WMMA_F32_16X16X4_F32.
// 8 waves, one 16-column output tile per wave, single f32 accumulator chain
// across all five weight matrices (K = 5*128 in chunks of 4).
//
// A-frag layout (16x4 f32): lanes 0-15 hold rows M=0-15 with {K=0,K=1};
// lanes 16-31 hold {K=2,K=3}.  Only row M=0 carries the vector, so only
// lanes 0 and 16 are non-zero.  To keep control flow uniform (EXEC must be
// all-ones for WMMA, and divergent LDS reads cost exec-mask churn), every
// lane reads LDS at its half's address (bank broadcast) and multiplies by a
// 0/1 mask -- no branches in the hot loop.
// B-frag (4x16): lane = column; lanes 0-15 hold {K=0,K=1}, lanes 16-31 {K=2,K=3}.
// D row 0 = VGPR0 of lanes 0-15 (C/D layout).
__global__ __launch_bounds__(256) void finalize_kernel(const float* __restrict__ acc,
                                                       const float* __restrict__ xi,
                                                       const float* __restrict__ W1,
                                                       const float* __restrict__ W2,
                                                       const float* __restrict__ W3,
                                                       const float* __restrict__ W4,
                                                       const float* __restrict__ W5,
                                                       const float* __restrict__ W6,
                                                       const float* __restrict__ W7,
                                                       const float* __restrict__ W8,
                                                       const float* __restrict__ W9,
                                                       const float* __restrict__ W10,
                                                       float* __restrict__ out) {
    __shared__ float sv[5 * 128];
    const int tid = threadIdx.x;

    if (tid < 128) {
        const int k = tid;
        sv[k] = acc[k];  // colsum(mu_N)
        sv[128 + k] = acc[128] * fmaxf(W3[k], 0.f) + acc[129] * fmaxf(-W3[k], 0.f);
        sv[256 + k] = acc[130] * fmaxf(W5[k], 0.f) + acc[131] * fmaxf(-W5[k], 0.f);
        sv[384 + k] = acc[132] * fmaxf(W7[k], 0.f) + acc[133] * fmaxf(-W7[k], 0.f);
        sv[512 + k] = acc[134] * fmaxf(W9[k], 0.f) + acc[135] * fmaxf(-W9[k], 0.f);
    }
    __syncthreads();

    const int wave    = tid >> 5;
    const int lane    = tid & 31;
    const int colBase = wave * 16;
    const int col     = colBase + (lane & 15);

    // A-fragment helpers: lanes 0-15 use K offset 0, lanes 16-31 use K offset 2;
    // only the first lane of each half (row M=0) contributes.
    const int   lsel  = (lane >> 4) << 1;                 // 0 or 2
    const float amask = ((lane & 15) == 0) ? 1.f : 0.f;   // lanes 0 and 16 only
    const int   kb0   = lsel;                             // B K-offset for this half

    const float* Ws[5] = {W1, W2, W4, W6, W8};

    v8f c = {0.f, 0.f, 0.f, 0.f, 0.f, 0.f, 0.f, 0.f};
    for (int m = 0; m < 5; ++m) {
        const float* __restrict__ W  = Ws[m];
        const float* __restrict__ vv = &sv[m * 128];
        #pragma unroll 4
        for (int kk = 0; kk < 32; ++kk) {
            const int k = kk * 4;
            v2f a;
            a.x = amask * vv[k + lsel];
            a.y = amask * vv[k + lsel + 1];
            v2f b;
            b.x = W[(size_t)(k + kb0) * 128 + col];
            b.y = W[(size_t)(k + kb0 + 1) * 128 + col];
            // D = A x B + C  (f32, 16x16x4)
            c = __builtin_amdgcn_wmma_f32_16x16x4_f32(
                    /*neg_a=*/false, a, /*neg_b=*/false, b,
                    /*c_mod=*/(short)0, c, /*reuse_a=*/false, /*reuse_b=*/false);
        }
    }

    if (lane < 16) {
        const int j = colBase + lane;
        float r = c[0] + xi[0] * W10[j] + xi[1] * W10[128 + j];
        out[j] = fmaxf(r, 0.f);
    }
}

extern "C" void kernel_launch(void* const* d_in, const int* in_sizes, int n_in,
                              void* d_out, int out_size, void* d_ws, size_t ws_size,
                              hipStream_t stream) {
    const float* xi  = (const float*)d_in[0];
    const float* muN = (const float*)d_in[1];
    const float* h   = (const float*)d_in[2];
    const float* hc  = (const float*)d_in[3];
    const float* s   = (const float*)d_in[4];
    const float* sc  = (const float*)d_in[5];
    const float* W1  = (const float*)d_in[6];
    const float* W2  = (const float*)d_in[7];
    const float* W3  = (const float*)d_in[8];
    const float* W4  = (const float*)d_in[9];
    const float* W5  = (const float*)d_in[10];
    const float* W6  = (const float*)d_in[11];
    const float* W7  = (const float*)d_in[12];
    const float* W8  = (const float*)d_in[13];
    const float* W9  = (const float*)d_in[14];
    const float* W10 = (const float*)d_in[15];
    float* out = (float*)d_out;
    float* ws  = (float*)d_ws;

    const int rows = in_sizes[1] / 128;
    const int nE   = in_sizes[2];

    zero_ws_kernel<<<1, 256, 0, stream>>>(ws);
    colsum_kernel<<<dim3(1024), 256, 0, stream>>>(muN, ws, rows);
    edge_sums_kernel<<<dim3(64, 4), 256, 0, stream>>>(h, hc, s, sc, nE, ws);
    finalize_kernel<<<1, 256, 0, stream>>>(ws, xi, W1, W2, W3, W4, W5, W6, W7,
                                           W8, W9, W10, out);
}